// Decoder_47691316855395
// MI455X (gfx1250) — compile-verified
//
#include <hip/hip_runtime.h>
#include <hip/hip_bf16.h>

typedef __attribute__((ext_vector_type(2))) float v2f;
typedef __attribute__((ext_vector_type(8))) float v8f;

#define MT 128          // batch tile
#define NT 128          // activation tile
#define KB 16           // K block per LDS stage
#define NSTAGE 3        // async pipeline depth (2 blocks always in flight)
#define XS_STRIDE 20    // padded x-tile row stride (floats): conflict-free A-frag reads
#define WS_STRIDE 136   // padded W-tile row stride (floats): conflict-free B-frag reads
#define X_FLOATS (MT * XS_STRIDE)           // 2560
#define W_FLOATS (KB * WS_STRIDE)           // 2176
#define STAGE_FLOATS (X_FLOATS + W_FLOATS)  // 4736
#define STAGE_BYTES (STAGE_FLOATS * 4)      // 18944 (x3 buffers = 56832 B LDS)

// C_i[m_base:+128, n_base:+128] = X[m, 0:K_i] * W_i[0:K_i, n], K_i=(i+1)*2048.
// f32 WMMA 16x16x4; W streamed once from HBM via 3-stage async global->LDS ring.
__global__ __launch_bounds__(256) void decoder_masked_gemm(
    const float* __restrict__ x,   // [256][12][2048]
    const float* __restrict__ W,   // [12][12][2048][768]
    float* __restrict__ out)       // [256][12][768]
{
    __shared__ __align__(16) float lds[NSTAGE * STAGE_FLOATS];

    const int t     = threadIdx.x;
    const int wave  = t >> 5;      // 0..7
    const int lane  = t & 31;
    const int lhalf = lane >> 4;   // selects K-pair (A/B frags) and M+8 (D)
    const int l16   = lane & 15;

    // Heaviest layers first; m-pairs of the same (i,n) adjacent for L2 W reuse.
    const int bid    = blockIdx.x;
    const int i      = 11 - (bid / 12);
    const int rem    = bid % 12;
    const int n_base = (rem >> 1) * NT;
    const int m_base = (rem & 1) * MT;

    const int K    = (i + 1) * 2048;
    const int NBLK = K / KB;       // >= 128

    // LDS byte base: low 32 bits of the generic pointer == workgroup LDS offset.
    const unsigned lds_base = (unsigned)(unsigned long long)(uintptr_t)&lds[0];

    // Per-lane constant pieces of the async-copy addressing (2 B128 each for x / W).
    unsigned x_voff[2], w_voff[2];           // per-lane global byte offsets (vs SGPR base)
    unsigned x_lda[2][NSTAGE], w_lda[2][NSTAGE];  // per-lane LDS byte addresses per buffer
    #pragma unroll
    for (int r = 0; r < 2; ++r) {
        int idx = t + 256 * r;
        {   // x tile: 128 rows x 16 f32 (4 float4/row)
            int row = idx >> 2, c4 = idx & 3;
            x_voff[r] = (unsigned)(((m_base + row) * 24576 + c4 * 4) * 4);
            unsigned l = lds_base + (unsigned)((row * XS_STRIDE + c4 * 4) * 4);
            #pragma unroll
            for (int s = 0; s < NSTAGE; ++s) x_lda[r][s] = l + s * STAGE_BYTES;
        }
        {   // W tile: 16 rows x 128 f32 (32 float4/row), row stride 768 in memory
            int row = idx >> 5, c4 = idx & 31;
            w_voff[r] = (unsigned)((row * 768 + c4 * 4) * 4);
            unsigned l = lds_base + (unsigned)(X_FLOATS * 4 + (row * WS_STRIDE + c4 * 4) * 4);
            #pragma unroll
            for (int s = 0; s < NSTAGE; ++s) w_lda[r][s] = l + s * STAGE_BYTES;
        }
    }

    // Issue one K-block (4 async B128 per wave -> ASYNCcnt += 4, retire in order).
    auto issue_block = [&](int kk, int buf) {
        const char* xsad = (const char*)x + (size_t)kk * 4;                      // + m*24576 elems per lane
        const char* wsad = (const char*)W +
            ((size_t)i * 18874368u + (size_t)kk * 768u + (size_t)n_base) * 4;    // + row*768 elems per lane
        #pragma unroll
        for (int r = 0; r < 2; ++r)
            asm volatile("global_load_async_to_lds_b128 %0, %1, %2"
                         :: "v"(x_lda[r][buf]), "v"(x_voff[r]), "s"(xsad) : "memory");
        #pragma unroll
        for (int r = 0; r < 2; ++r)
            asm volatile("global_load_async_to_lds_b128 %0, %1, %2"
                         :: "v"(w_lda[r][buf]), "v"(w_voff[r]), "s"(wsad) : "memory");
    };

    v8f acc[8] = {};  // 8 n-subtiles of 16x16 f32 per wave

    // Prologue: two blocks in flight (NBLK >= 128 always).
    issue_block(0, 0);
    issue_block(KB, 1);

    int cur = 0;  // buffer holding block b
    for (int b = 0; b < NBLK; ++b) {
        if (b + 2 < NBLK) {
            // refill buffer (b+2)%3 == (cur+2)%3; its previous contents (block b-1)
            // were consumed before iteration b's entry barrier from iteration b-1.
            issue_block((b + 2) * KB, cur == 0 ? 2 : cur - 1);
            asm volatile("s_wait_asynccnt 0x8" ::: "memory");  // block b's 4 loads done
        } else if (b + 1 < NBLK) {
            asm volatile("s_wait_asynccnt 0x4" ::: "memory");
        } else {
            asm volatile("s_wait_asynccnt 0x0" ::: "memory");
        }
        __syncthreads();  // all waves' async writes for block b visible

        const float* xs = lds + cur * STAGE_FLOATS;
        const float* ws = xs + X_FLOATS;

        #pragma unroll
        for (int ks = 0; ks < KB / 4; ++ks) {
            const int kb = ks * 4 + 2 * lhalf;
            v2f a;
            const float* xp = xs + (wave * 16 + l16) * XS_STRIDE + kb;
            a.x = xp[0];
            a.y = xp[1];
            #pragma unroll
            for (int ns = 0; ns < 8; ++ns) {
                v2f bb;
                const float* wp = ws + kb * WS_STRIDE + ns * 16 + l16;
                bb.x = wp[0];
                bb.y = wp[WS_STRIDE];
                acc[ns] = __builtin_amdgcn_wmma_f32_16x16x4_f32(
                    false, a, false, bb, (short)0, acc[ns], false, false);
            }
        }
        __syncthreads();  // buf `cur` free for refill at iteration b+1
        cur = (cur == 2) ? 0 : cur + 1;
    }

    // D layout: vgpr v -> M = v + 8*lhalf, N = l16 (16-lane contiguous segments)
    #pragma unroll
    for (int ns = 0; ns < 8; ++ns) {
        #pragma unroll
        for (int v = 0; v < 8; ++v) {
            const int bb = m_base + wave * 16 + v + 8 * lhalf;
            const int a_col = n_base + ns * 16 + l16;
            out[((size_t)bb * 12 + i) * 768 + a_col] = acc[ns][v];
        }
    }
}

extern "C" void kernel_launch(void* const* d_in, const int* in_sizes, int n_in,
                              void* d_out, int out_size, void* d_ws, size_t ws_size,
                              hipStream_t stream) {
    (void)in_sizes; (void)n_in; (void)d_ws; (void)ws_size; (void)out_size;
    const float* x = (const float*)d_in[0];  // [256,12,2048]
    const float* W = (const float*)d_in[1];  // [12,12,2048,768]
    float* out = (float*)d_out;              // [256,12,768]

    // 12 layers x 6 n-tiles x 2 m-tiles = 144 workgroups of 8 wave32s
    decoder_masked_gemm<<<dim3(144), dim3(256), 0, stream>>>(x, W, out);
}